// GR_78434692759819
// MI455X (gfx1250) — compile-verified
//
#include <hip/hip_runtime.h>
#include <hip/hip_bf16.h>

typedef __attribute__((ext_vector_type(2))) float v2f;
typedef __attribute__((ext_vector_type(8))) float v8f;

#define N 4096
#define NTILE_C 1024   // 4096/4 tiles along k
#define NTILE_R 256    // 4096/16 row blocks
#define ELEMS 16777216u

// ---------------------------------------------------------------------------
// Build K = exp(-20*M) directly into two WMMA-A-swizzled layouts:
//   K_swz : A tiles of K    (rows = i, k-dim = j)   used for  x = K v
//   KT_swz: A tiles of K^T  (rows = j, k-dim = i)   used for  y = K^T u
// A-tile layout (16x4, f32): lane l holds rows M=l&15, cols (l>>4)*2 + {0,1};
// tile (tr,tc) stored at ((tr*1024)+tc)*64, element at lane*2+e.
// ---------------------------------------------------------------------------
__global__ __launch_bounds__(256) void GR_init_swz(const float* __restrict__ Mm,
                                                   float* __restrict__ Kswz,
                                                   float* __restrict__ KTswz) {
    unsigned idx = blockIdx.x * 256u + threadIdx.x;     // 65536 blocks -> 16.7M
    unsigned i = idx >> 12;
    unsigned j = idx & 4095u;
    float k = expf(-20.0f * Mm[idx]);
    unsigned laneK  = (i & 15u) | (((j >> 1) & 1u) << 4);
    unsigned offK   = ((i >> 4) * 1024u + (j >> 2)) * 64u + (laneK << 1) + (j & 1u);
    unsigned laneKT = (j & 15u) | (((i >> 1) & 1u) << 4);
    unsigned offKT  = ((j >> 4) * 1024u + (i >> 2)) * 64u + (laneKT << 1) + (i & 1u);
    Kswz[offK]   = k;
    KTswz[offKT] = k;
}

// a[i] = sum_j group[i][j] : one 256-thread block per row, coalesced
__global__ __launch_bounds__(256) void GR_rowsum(const float* __restrict__ g,
                                                 float* __restrict__ a) {
    __shared__ float s[256];
    const float* gr = g + (size_t)blockIdx.x * N;
    float acc = 0.0f;
    for (int j = threadIdx.x; j < N; j += 256) acc += gr[j];
    s[threadIdx.x] = acc;
    __syncthreads();
    for (int st = 128; st > 0; st >>= 1) {
        if (threadIdx.x < st) s[threadIdx.x] += s[threadIdx.x + st];
        __syncthreads();
    }
    if (threadIdx.x == 0) a[blockIdx.x] = s[0];
}

// b[j] = sum_i group[i][j] : thread per column, coalesced across threads
__global__ __launch_bounds__(256) void GR_colsum(const float* __restrict__ g,
                                                 float* __restrict__ b) {
    int c = blockIdx.x * 256 + threadIdx.x;
    float acc = 0.0f;
    for (int i = 0; i < N; ++i) acc += g[(size_t)i * N + c];
    b[c] = acc;
}

__global__ void GR_vecinit(float* __restrict__ u, float* __restrict__ v,
                           int* __restrict__ done) {
    int i = blockIdx.x * 256 + threadIdx.x;
    u[i] = 1.0f / 4096.0f;
    v[i] = 0.0f;
    if (i == 0) *done = 0;
}

// ---------------------------------------------------------------------------
// WMMA matvec over a swizzled A matrix. Each block owns one 16-row output
// block; 8 waves split the 4096-long k dimension (128 tiles each), the
// vector is staged in LDS, partial v8f accumulators are reduced via LDS.
// mode 0: outv[r] = numer[r] / (s_r + 1e-16)           (u / v update)
// mode 1: errpart[block] = sum_r |vcur[r]*s_r - bref[r]|  (error check)
// If *done, the kernel is a uniform no-op (frozen iterates).
// ---------------------------------------------------------------------------
__global__ __launch_bounds__(256) void GR_matvec(const float* __restrict__ Aswz,
                                                 const float* __restrict__ x,
                                                 const float* __restrict__ numer,
                                                 float* __restrict__ outv,
                                                 const float* __restrict__ bref,
                                                 const float* __restrict__ vcur,
                                                 float* __restrict__ errpart,
                                                 const int* __restrict__ done,
                                                 int mode) {
    if (*done) return;                    // uniform across the whole grid
    __shared__ float xs[N];
    __shared__ float red[8 * 16];
    int tid = threadIdx.x;
    for (int k = tid; k < N; k += 256) xs[k] = x[k];
    __syncthreads();

    int wave = tid >> 5;                  // 0..7
    int lane = tid & 31;
    int rb   = blockIdx.x;                // 0..255 row blocks
    const float* base = Aswz + ((size_t)rb * NTILE_C + (size_t)wave * 128) * 64 + lane * 2;
    int koff  = (lane >> 4) * 2;          // B layout: lanes>=16 hold k rows 2,3
    int kbase = wave * 512;               // wave*128*4

    v8f c = {0.f, 0.f, 0.f, 0.f, 0.f, 0.f, 0.f, 0.f};
#pragma unroll 4
    for (int t = 0; t < 128; ++t) {
        v2f av = *(const v2f*)(base + t * 64);            // coalesced 256B/wave
        v2f bv = *(const v2f*)(&xs[kbase + t * 4 + koff]); // vector slice (all cols equal)
        c = __builtin_amdgcn_wmma_f32_16x16x4_f32(
                /*neg_a=*/false, av, /*neg_b=*/false, bv,
                /*c_mod=*/(short)0, c, /*reuse_a=*/false, /*reuse_b=*/false);
    }

    // D layout: lane 0 holds rows 0..7 in c[0..7]; lane 16 holds rows 8..15.
    if (lane == 0) {
#pragma unroll
        for (int r = 0; r < 8; ++r) red[wave * 16 + r] = c[r];
    } else if (lane == 16) {
#pragma unroll
        for (int r = 0; r < 8; ++r) red[wave * 16 + 8 + r] = c[r];
    }
    __syncthreads();

    if (mode == 0) {
        if (tid < 16) {
            float s = 0.0f;
#pragma unroll
            for (int w = 0; w < 8; ++w) s += red[w * 16 + tid];
            int row = rb * 16 + tid;
            outv[row] = numer[row] / (s + 1e-16f);
        }
    } else {
        if (tid < 16) {
            float s = 0.0f;
#pragma unroll
            for (int w = 0; w < 8; ++w) s += red[w * 16 + tid];
            int row = rb * 16 + tid;
            red[tid] = fabsf(vcur[row] * s - bref[row]);
        }
        __syncthreads();
        if (tid == 0) {
            float s = 0.0f;
            for (int j = 0; j < 16; ++j) s += red[j];
            errpart[rb] = s;              // no atomics -> deterministic
        }
    }
}

// err = sum of 256 block partials (fixed order); flip done if converged.
__global__ void GR_finalize_err(const float* __restrict__ errpart,
                                int* __restrict__ done) {
    float s = 0.0f;
    for (int i = 0; i < 256; ++i) s += errpart[i];
    if (s <= 0.005f) *done = 1;
}

// loss partials: recompute K elementwise (saves storing plain K),
// transp = clamp(u_i*K_ij*v_j, 1e-6), KL-style sum. Fixed-order per thread.
__global__ __launch_bounds__(256) void GR_loss(const float* __restrict__ Mm,
                                               const float* __restrict__ g,
                                               const float* __restrict__ u,
                                               const float* __restrict__ v,
                                               float* __restrict__ part) {
    __shared__ float s[256];
    unsigned stride = gridDim.x * 256u;
    float acc = 0.0f;
    for (unsigned p = blockIdx.x * 256u + threadIdx.x; p < ELEMS; p += stride) {
        unsigned i = p >> 12, j = p & 4095u;
        float k = expf(-20.0f * Mm[p]);
        float t = fmaxf(u[i] * k * v[j], 1e-6f);
        float gg = g[p];
        acc += gg * (logf(gg) - logf(t) - 1.0f) + t;
    }
    s[threadIdx.x] = acc;
    __syncthreads();
    for (int st = 128; st > 0; st >>= 1) {
        if (threadIdx.x < st) s[threadIdx.x] += s[threadIdx.x + st];
        __syncthreads();
    }
    if (threadIdx.x == 0) part[blockIdx.x] = s[0];
}

__global__ __launch_bounds__(256) void GR_loss_fin(const float* __restrict__ part,
                                                   float* __restrict__ out) {
    __shared__ float s[256];
    float acc = 0.0f;
    for (int i = threadIdx.x; i < 16384; i += 256) acc += part[i];
    s[threadIdx.x] = acc;
    __syncthreads();
    for (int st = 128; st > 0; st >>= 1) {
        if (threadIdx.x < st) s[threadIdx.x] += s[threadIdx.x + st];
        __syncthreads();
    }
    if (threadIdx.x == 0) out[0] = s[0];
}

extern "C" void kernel_launch(void* const* d_in, const int* in_sizes, int n_in,
                              void* d_out, int out_size, void* d_ws, size_t ws_size,
                              hipStream_t stream) {
    const float* M     = (const float*)d_in[0];
    const float* group = (const float*)d_in[1];
    float* out = (float*)d_out;

    float* wsf      = (float*)d_ws;
    float* Kswz     = wsf;                    // 16.7M
    float* KTswz    = Kswz  + ELEMS;          // 16.7M
    float* a        = KTswz + ELEMS;          // 4096
    float* b        = a + N;                  // 4096
    float* u        = b + N;                  // 4096
    float* v        = u + N;                  // 4096
    float* errpart  = v + N;                  // 256
    float* losspart = errpart + 256;          // 16384
    int*   done     = (int*)(losspart + 16384);

    GR_init_swz<<<ELEMS / 256, 256, 0, stream>>>(M, Kswz, KTswz);
    GR_rowsum<<<N, 256, 0, stream>>>(group, a);
    GR_colsum<<<N / 256, 256, 0, stream>>>(group, b);
    GR_vecinit<<<N / 256, 256, 0, stream>>>(u, v, done);

    for (int it = 0; it < 200; ++it) {
        // v = b / (K^T u + eps)
        GR_matvec<<<NTILE_R, 256, 0, stream>>>(KTswz, u, b, v,
                                               nullptr, nullptr, nullptr, done, 0);
        // u = a / (K v + eps)
        GR_matvec<<<NTILE_R, 256, 0, stream>>>(Kswz, v, a, u,
                                               nullptr, nullptr, nullptr, done, 0);
        if (it % 50 == 0) {                   // (i+1)%50==1 check points
            GR_matvec<<<NTILE_R, 256, 0, stream>>>(KTswz, u, nullptr, nullptr,
                                                   b, v, errpart, done, 1);
            GR_finalize_err<<<1, 1, 0, stream>>>(errpart, done);
        }
    }

    GR_loss<<<16384, 256, 0, stream>>>(M, group, u, v, losspart);
    GR_loss_fin<<<1, 256, 0, stream>>>(losspart, out);
}